// HyperbolicTreeLayer_86303072846399
// MI455X (gfx1250) — compile-verified
//
#include <hip/hip_runtime.h>
#include <hip/hip_bf16.h>
#include <math.h>

// ---------------------------------------------------------------------------
// HyperbolicTreeLayer for MI455X (gfx1250, wave32, WMMA)
//
// Pass 1 (edges): log-map v per edge (one edge per lane), scorer MLP via
//                 V_WMMA_F32_16X16X4_F32 (Ht = W1t[16x4] x Vt[4x16], bias via
//                 C accumulator), gelu + dot(W2) via 8 FMAs + one
//                 ds_swizzle_b32 SWAPX16 reduction. Store score (and v if the
//                 workspace is big enough), atomicMax per target node.
// Pass 2 (edges): load (or recompute) v, ex = exp(score - max),
//                 atomicAdd Σex and Σex*v per node.
// Pass 3 (nodes): m = Σex*v / max(Σex,1e-15), depth-gated rotate/scale,
//                 exp-map + Möbius add -> out.
// ---------------------------------------------------------------------------

typedef __attribute__((ext_vector_type(2))) float v2f;
typedef __attribute__((ext_vector_type(8))) float v8f;

#define EPSF 1e-15f

// Single-instruction half-wave swap (wave32): ds_swizzle_b32 SWAPX16
// offset = {grp32 mode, xor_mask=0x10, or_mask=0, and_mask=0x1f} = 0x401F
__device__ __forceinline__ float swap16(float v) {
    return __int_as_float(__builtin_amdgcn_ds_swizzle(__float_as_int(v), 0x401F));
}

__device__ __forceinline__ unsigned ordf(float f) {
    unsigned u = __float_as_uint(f);
    return ((int)u < 0) ? ~u : (u ^ 0x80000000u);
}
__device__ __forceinline__ float unordf(unsigned u) {
    unsigned v = ((int)u < 0) ? (u ^ 0x80000000u) : ~u;
    return __uint_as_float(v);
}

__device__ __forceinline__ float gelu_exact(float z) {
    return 0.5f * z * (1.0f + erff(z * 0.70710678118654752440f));
}

// log_{xi}(xj) on the Poincare disk, c = 1 (2-D); x accessed as float2
__device__ __forceinline__ void log_map(const float2* __restrict__ x,
                                        int r, int c, float& vx, float& vy) {
    float2 xi = x[r];
    float2 xj = x[c];
    float ax = -xi.x, ay = -xi.y;                    // a = -xi, b = xj
    float ab = ax * xj.x + ay * xj.y;
    float aa = ax * ax + ay * ay;
    float bb = xj.x * xj.x + xj.y * xj.y;
    float f1 = 1.0f + 2.0f * ab + bb;
    float f2 = 1.0f - aa;
    float nx = f1 * ax + f2 * xj.x;
    float ny = f1 * ay + f2 * xj.y;
    float den = fmaxf(1.0f + 2.0f * ab + aa * bb, EPSF);
    float inv = 1.0f / den;
    float ux = nx * inv, uy = ny * inv;
    float un = fmaxf(sqrtf(ux * ux + uy * uy), EPSF);
    float li = fmaxf(1.0f - (xi.x * xi.x + xi.y * xi.y), EPSF); // = 2/(sc*lam_i)
    float t = atanhf(fminf(un, 1.0f - 1e-5f));
    float s = li * t / un;
    vx = s * ux;
    vy = s * uy;
}

__global__ void init_nodes_kernel(unsigned* __restrict__ nmax,
                                  float* __restrict__ denom,
                                  float* __restrict__ mx,
                                  float* __restrict__ my, int N) {
    int i = blockIdx.x * blockDim.x + threadIdx.x;
    if (i < N) {
        nmax[i] = 0u;      // below ordf(-inf)
        denom[i] = 0.0f;
        mx[i] = 0.0f;
        my[i] = 0.0f;
    }
}

// Pass 1: scores via WMMA MLP + per-node atomic max.
// Every wave runs full-EXEC through the WMMAs; tail lanes clamp their loads
// and predicate only their global stores/atomics.
template <bool STOREV>
__global__ void __launch_bounds__(256)
edge_score_kernel(const float2* __restrict__ x,
                  const int* __restrict__ rowi,
                  const int* __restrict__ coli,
                  const float* __restrict__ W1,   // [2,16] row-major
                  const float* __restrict__ b1,   // [16]
                  const float* __restrict__ W2,   // [16]
                  float* __restrict__ scores,
                  float* __restrict__ vxbuf,
                  float* __restrict__ vybuf,
                  unsigned* __restrict__ nmax,
                  int E) {
    int e = blockIdx.x * blockDim.x + threadIdx.x;
    int lane = threadIdx.x & 31;
    bool valid = (e < E);
    int ee = valid ? e : (E - 1);

    // prefetch the edge-index stream ahead (global_prefetch_b8)
    if (ee + 8192 < E) {
        __builtin_prefetch(&rowi[ee + 8192], 0, 1);
        __builtin_prefetch(&coli[ee + 8192], 0, 1);
    }

    int r = rowi[ee];
    int c = coli[ee];

    float vx, vy;
    log_map(x, r, c, vx, vy);

    bool lo = (lane < 16);
    int  ln = lane & 15;
    int  hb = lo ? 0 : 8;        // hidden-row base held by this half-wave

    // A = W1^T, 16x4 (K=2,3 zero padding on high half-wave)
    v2f A;
    A.x = lo ? W1[ln]      : 0.0f;   // W1[0, ln]
    A.y = lo ? W1[16 + ln] : 0.0f;   // W1[1, ln]

    // C = bias broadcast: C[m, n] = b1[m]
    v8f C;
    float w2l[8];
#pragma unroll
    for (int k = 0; k < 8; ++k) {
        C[k]   = b1[k + hb];
        w2l[k] = W2[k + hb];
    }

    // WMMA #1: edges [wavebase .. wavebase+15]  (B = V^T, lanes 0-15)
    v2f B1;
    B1.x = lo ? vx : 0.0f;
    B1.y = lo ? vy : 0.0f;
    v8f D1 = __builtin_amdgcn_wmma_f32_16x16x4_f32(
        false, A, false, B1, (short)0, C, false, false);

    // WMMA #2: edges [wavebase+16 .. wavebase+31]
    float vx2 = swap16(vx);
    float vy2 = swap16(vy);
    v2f B2;
    B2.x = lo ? vx2 : 0.0f;
    B2.y = lo ? vy2 : 0.0f;
    v8f D2 = __builtin_amdgcn_wmma_f32_16x16x4_f32(
        false, A, false, B2, (short)0, C, false, false);

    // gelu + dot with W2: lane n holds H[hb..hb+7, n]; partial then swap16 add.
    float p1 = 0.0f, p2 = 0.0f;
#pragma unroll
    for (int k = 0; k < 8; ++k) {
        p1 += gelu_exact(D1[k]) * w2l[k];
        p2 += gelu_exact(D2[k]) * w2l[k];
    }
    float s1 = p1 + swap16(p1);   // score of edge wavebase + (lane&15)
    float s2 = p2 + swap16(p2);   // score of edge wavebase + 16 + (lane&15)
    float score = lo ? s1 : s2;   // lane L owns edge wavebase + L

    if (valid) {
        scores[e] = score;
        if (STOREV) {
            vxbuf[e] = vx;
            vybuf[e] = vy;
        }
        atomicMax(&nmax[r], ordf(score));
    }
}

// Pass 2: ex = exp(score - max), accumulate Σex and Σex*v per node.
template <bool STOREV>
__global__ void __launch_bounds__(256)
edge_accum_kernel(const float2* __restrict__ x,
                  const int* __restrict__ rowi,
                  const int* __restrict__ coli,
                  const float* __restrict__ scores,
                  const float* __restrict__ vxbuf,
                  const float* __restrict__ vybuf,
                  const unsigned* __restrict__ nmax,
                  float* __restrict__ denom,
                  float* __restrict__ mx,
                  float* __restrict__ my,
                  int E) {
    int e = blockIdx.x * blockDim.x + threadIdx.x;
    if (e >= E) return;
    if (e + 8192 < E) {
        __builtin_prefetch(&rowi[e + 8192], 0, 1);
        __builtin_prefetch(&scores[e + 8192], 0, 1);
    }
    int r = rowi[e];
    float vx, vy;
    if (STOREV) {
        vx = vxbuf[e];
        vy = vybuf[e];
    } else {
        int c = coli[e];
        log_map(x, r, c, vx, vy);
    }
    float m = unordf(nmax[r]);
    float ex = expf(scores[e] - m);
    atomicAdd(&denom[r], ex);
    atomicAdd(&mx[r], ex * vx);
    atomicAdd(&my[r], ex * vy);
}

// Pass 3: per-node mixer + exp-map + Möbius add.
__global__ void __launch_bounds__(256)
node_finish_kernel(const float2* __restrict__ x,
                   const int* __restrict__ depth,
                   const float* __restrict__ dscale,   // [512]
                   const float* __restrict__ dtheta,   // [512]
                   const float* __restrict__ eta,
                   const float* __restrict__ denom,
                   const float* __restrict__ mx,
                   const float* __restrict__ my,
                   float2* __restrict__ out,
                   int N) {
    int i = blockIdx.x * blockDim.x + threadIdx.x;
    if (i >= N) return;

    float den = fmaxf(denom[i], 1e-15f);
    float m0 = mx[i] / den;
    float m1 = my[i] / den;

    int d = depth[i];
    d = d < 0 ? 0 : (d > 511 ? 511 : d);
    float k   = dscale[d];
    float ang = dtheta[d];
    float ca = cosf(ang), sa = sinf(ang);
    float r0 = k * (ca * m0 - sa * m1);
    float r1 = k * (sa * m0 + ca * m1);

    float ev = eta[0];
    float w0 = ev * r0, w1 = ev * r1;
    float wn = fmaxf(sqrtf(w0 * w0 + w1 * w1), EPSF);

    float2 xi = x[i];
    float li = fmaxf(1.0f - (xi.x * xi.x + xi.y * xi.y), EPSF);
    float th = tanhf(wn / li);                 // tanh(sc*lam_x*wn/2), lam=2/li
    float s0 = th * w0 / wn;
    float s1v = th * w1 / wn;

    // mobius_add(x, second), c = 1
    float ab = xi.x * s0 + xi.y * s1v;
    float aa = xi.x * xi.x + xi.y * xi.y;
    float bb = s0 * s0 + s1v * s1v;
    float f1 = 1.0f + 2.0f * ab + bb;
    float f2 = 1.0f - aa;
    float n0 = f1 * xi.x + f2 * s0;
    float n1 = f1 * xi.y + f2 * s1v;
    float dd = fmaxf(1.0f + 2.0f * ab + aa * bb, EPSF);
    float2 o;
    o.x = n0 / dd;
    o.y = n1 / dd;
    out[i] = o;
}

extern "C" void kernel_launch(void* const* d_in, const int* in_sizes, int n_in,
                              void* d_out, int out_size, void* d_ws, size_t ws_size,
                              hipStream_t stream) {
    const float2* x     = (const float2*)d_in[0];  // [N,2]
    const int*   edge   = (const int*)  d_in[1];   // [2,E]
    const int*   depth  = (const int*)  d_in[2];   // [N]
    const float* W1     = (const float*)d_in[3];   // [2,16]
    const float* b1     = (const float*)d_in[4];   // [16]
    const float* W2     = (const float*)d_in[5];   // [16,1]
    const float* eta    = (const float*)d_in[6];   // [1]
    const float* dscale = (const float*)d_in[7];   // [512,1]
    const float* dtheta = (const float*)d_in[8];   // [512,1]

    const int N = in_sizes[0] / 2;
    const int E = in_sizes[1] / 2;
    const int* rowi = edge;
    const int* coli = edge + E;

    // Workspace carve. Big layout: scores[E] vx[E] vy[E] nmax[N] den[N] mx[N] my[N]
    // Small layout (v recomputed in pass 2): scores[E] nmax[N] den[N] mx[N] my[N]
    const size_t need_big = sizeof(float) * ((size_t)E * 3 + (size_t)N * 4);
    const bool storev = (ws_size >= need_big);

    float*    scores = (float*)d_ws;
    float*    vxbuf  = storev ? (scores + E) : nullptr;
    float*    vybuf  = storev ? (scores + 2 * (size_t)E) : nullptr;
    float*    ntop   = storev ? (scores + 3 * (size_t)E) : (scores + E);
    unsigned* nmax   = (unsigned*)ntop;
    float*    denom  = (float*)(nmax + N);
    float*    mx     = denom + N;
    float*    my     = mx + N;

    const int TB = 256;
    int nblkN = (N + TB - 1) / TB;
    int nblkE = (E + TB - 1) / TB;

    init_nodes_kernel<<<nblkN, TB, 0, stream>>>(nmax, denom, mx, my, N);
    if (storev) {
        edge_score_kernel<true><<<nblkE, TB, 0, stream>>>(
            x, rowi, coli, W1, b1, W2, scores, vxbuf, vybuf, nmax, E);
        edge_accum_kernel<true><<<nblkE, TB, 0, stream>>>(
            x, rowi, coli, scores, vxbuf, vybuf, nmax, denom, mx, my, E);
    } else {
        edge_score_kernel<false><<<nblkE, TB, 0, stream>>>(
            x, rowi, coli, W1, b1, W2, scores, nullptr, nullptr, nmax, E);
        edge_accum_kernel<false><<<nblkE, TB, 0, stream>>>(
            x, rowi, coli, scores, nullptr, nullptr, nmax, denom, mx, my, E);
    }
    node_finish_kernel<<<nblkN, TB, 0, stream>>>(x, depth, dscale, dtheta, eta,
                                                 denom, mx, my, (float2*)d_out, N);
}